// RecurrentGNN_41979010351135
// MI455X (gfx1250) — compile-verified
//
#include <hip/hip_runtime.h>
#include <hip/hip_bf16.h>

// N=100000, E=1600000, IN=256, H=128, OUT=16, L=3
#define HID 128
#define INF 256
#define OUTF 16

typedef float v2f __attribute__((ext_vector_type(2)));
typedef float v8f __attribute__((ext_vector_type(8)));

__device__ __forceinline__ float sigmf(float x) {
    return 1.0f / (1.0f + __expf(-x));
}

// CDNA5 async global->LDS DMA (bypasses VGPRs, tracked by ASYNCcnt).
// lds_off: workgroup-relative LDS byte offset (low 32 bits of generic addr).
__device__ __forceinline__ void async_g2l_b128(unsigned int lds_off,
                                               const float* g) {
    asm volatile("global_load_async_to_lds_b128 %0, %1, off"
                 :: "v"(lds_off), "v"((unsigned long long)(uintptr_t)g)
                 : "memory");
}
__device__ __forceinline__ void wait_async0() {
    asm volatile("s_wait_asynccnt 0x0" ::: "memory");
}

// ---------------------------------------------------------------------------
// Generic C[M,Nc] = A[M,K] @ W[Nc,K]^T (+ bias).  fp32 WMMA 16x16x4.
// block = 256 threads (8 waves). grid = (M/16, ceil(Nc/128)).
// Wave w computes the 16x16 tile at cols blockIdx.y*128 + w*16.
// A tile (16 x K) DMA'd into LDS (async-to-LDS) with +4-float row padding
// (K multiple of 64 -> conflict-free ds_load_2addr fragment reads).
// ---------------------------------------------------------------------------
__global__ void gemm_nt_wmma(const float* __restrict__ A,
                             const float* __restrict__ W,
                             const float* __restrict__ bias,
                             float* __restrict__ C,
                             int M, int Nc, int K) {
    __shared__ float As[16 * (256 + 4)];  // supports K <= 256
    const int KP = K + 4;
    const int tid = threadIdx.x;
    const int row0 = blockIdx.x * 16;

    // Async DMA of the contiguous 16xK A tile into padded LDS rows.
    {
        const int k4 = K >> 2;  // float4 chunks per row
        const unsigned int lds_base = (unsigned int)(uintptr_t)&As[0];
        const float* gbase = A + (size_t)row0 * K;
        for (int i4 = tid; i4 < 16 * k4; i4 += 256) {
            int r = i4 / k4, c4 = i4 - r * k4;
            async_g2l_b128(lds_base + (unsigned int)(r * KP + c4 * 4) * 4u,
                           gbase + (size_t)i4 * 4);
        }
        wait_async0();
    }
    __syncthreads();

    const int wv = tid >> 5;
    const int lane = tid & 31;
    const int n0 = blockIdx.y * 128 + wv * 16;
    if (n0 >= Nc) return;  // wave-uniform

    const int m = lane & 15;        // A row within tile / C col within tile
    const int half = lane >> 4;     // 0 or 1
    const int klo = half * 2;       // K sub-offset for A/B fragments

    v8f acc;
    {
        float bv = bias ? bias[n0 + m] : 0.0f;
#pragma unroll
        for (int v = 0; v < 8; ++v) acc[v] = bv;
    }

    const float* Ar = As + m * KP;
    const float* Wr = W + (size_t)(n0 + m) * K;  // row (n0+n) of W, n = lane&15

    for (int k = 0; k < K; k += 4) {
        v2f a; a[0] = Ar[k + klo]; a[1] = Ar[k + klo + 1];
        v2f b; b[0] = Wr[k + klo]; b[1] = Wr[k + klo + 1];
        acc = __builtin_amdgcn_wmma_f32_16x16x4_f32(false, a, false, b,
                                                    (short)0, acc, false, false);
    }

    const int rbase = row0 + half * 8;
#pragma unroll
    for (int v = 0; v < 8; ++v)
        C[(size_t)(rbase + v) * Nc + n0 + m] = acc[v];
}

// ---------------------------------------------------------------------------
// Fused single-timestep LSTM (h0=c0=0): h = sig(o)*tanh(sig(i)*tanh(g))
// gates = h_in @ w_ih^T + (b_ih + b_hh); f-gate dead (f*c0 == 0) -> skipped.
// block = 256 threads (8 waves), grid = M/16. Wave w owns hidden cols
// j0 = w*16 across the three live gates (col bases q*128 + j0, q in {0,2,3}).
// ---------------------------------------------------------------------------
__global__ void lstm_fused_wmma(const float* __restrict__ hin,
                                const float* __restrict__ w_ih,
                                const float* __restrict__ b_ih,
                                const float* __restrict__ b_hh,
                                float* __restrict__ hout) {
    const int K = HID, KP = HID + 4;
    __shared__ float As[16 * (HID + 4)];
    const int tid = threadIdx.x;
    const int row0 = blockIdx.x * 16;

    {
        const int k4 = K >> 2;
        const unsigned int lds_base = (unsigned int)(uintptr_t)&As[0];
        const float* gbase = hin + (size_t)row0 * K;
        for (int i4 = tid; i4 < 16 * k4; i4 += 256) {
            int r = i4 / k4, c4 = i4 - r * k4;
            async_g2l_b128(lds_base + (unsigned int)(r * KP + c4 * 4) * 4u,
                           gbase + (size_t)i4 * 4);
        }
        wait_async0();
    }
    __syncthreads();

    const int wv = tid >> 5;
    const int lane = tid & 31;
    const int m = lane & 15;
    const int half = lane >> 4;
    const int klo = half * 2;
    const int j0 = wv * 16;

    const int gq[3] = {0, 2, 3};  // i, g, o
    v8f acc[3];
    const float* Wr[3];
#pragma unroll
    for (int q = 0; q < 3; ++q) {
        int col = gq[q] * HID + j0 + m;
        float bv = b_ih[col] + b_hh[col];
#pragma unroll
        for (int v = 0; v < 8; ++v) acc[q][v] = bv;
        Wr[q] = w_ih + (size_t)col * K;
    }

    const float* Ar = As + m * KP;
    for (int k = 0; k < K; k += 4) {
        v2f a; a[0] = Ar[k + klo]; a[1] = Ar[k + klo + 1];
#pragma unroll
        for (int q = 0; q < 3; ++q) {
            v2f b; b[0] = Wr[q][k + klo]; b[1] = Wr[q][k + klo + 1];
            acc[q] = __builtin_amdgcn_wmma_f32_16x16x4_f32(false, a, false, b,
                                                           (short)0, acc[q],
                                                           false, false);
        }
    }

    const int rbase = row0 + half * 8;
#pragma unroll
    for (int v = 0; v < 8; ++v) {
        float ig = sigmf(acc[0][v]);
        float gg = tanhf(acc[1][v]);
        float og = sigmf(acc[2][v]);
        float c = ig * gg;
        hout[(size_t)(rbase + v) * HID + j0 + m] = og * tanhf(c);
    }
}

// ---------------------------------------------------------------------------
// Edge-degree accumulation (dst side; self-loop folded into dinv kernel).
// ---------------------------------------------------------------------------
__global__ void deg_kernel(const int* __restrict__ ei, float* __restrict__ deg,
                           int E) {
    int e = blockIdx.x * blockDim.x + threadIdx.x;
    if (e < E) unsafeAtomicAdd(&deg[ei[E + e]], 1.0f);
}

__global__ void dinv_kernel(float* __restrict__ deg, int n) {
    int i = blockIdx.x * blockDim.x + threadIdx.x;
    if (i < n) deg[i] = rsqrtf(deg[i] + 1.0f);  // +1 = self loop; always > 0
}

// ---------------------------------------------------------------------------
// One wave per edge: agg[dst,:] += xw[src,:] * dinv[src]*dinv[dst]
// lane handles 4 consecutive floats of the 128-float row.
// xw (51 MB) and agg (51 MB) both fit in the 192 MB L2 -> gathers and
// atomics run at L2 rate.
// ---------------------------------------------------------------------------
__global__ void scatter_edges(const int* __restrict__ ei,
                              const float* __restrict__ xw,
                              const float* __restrict__ dinv,
                              float* __restrict__ agg, int E) {
    int gw = (blockIdx.x * blockDim.x + threadIdx.x) >> 5;
    int lane = threadIdx.x & 31;
    if (gw >= E) return;
    int src = ei[gw];
    int dst = ei[E + gw];
    float norm = dinv[src] * dinv[dst];
    float4 v = reinterpret_cast<const float4*>(xw + (size_t)src * HID)[lane];
    float* out = agg + (size_t)dst * HID + lane * 4;
    unsafeAtomicAdd(out + 0, v.x * norm);
    unsafeAtomicAdd(out + 1, v.y * norm);
    unsafeAtomicAdd(out + 2, v.z * norm);
    unsafeAtomicAdd(out + 3, v.w * norm);
}

// ---------------------------------------------------------------------------
// agg = relu(agg + xw * dinv^2 + gcn_b)   (self-loop term + bias, in place)
// ---------------------------------------------------------------------------
__global__ void selfloop_bias_relu(float* __restrict__ agg,
                                   const float* __restrict__ xw,
                                   const float* __restrict__ dinv,
                                   const float* __restrict__ gcn_b, int n) {
    int idx = blockIdx.x * blockDim.x + threadIdx.x;
    if (idx >= n * HID) return;
    int node = idx >> 7;
    int j = idx & (HID - 1);
    float di = dinv[node];
    float v = agg[idx] + xw[idx] * di * di + gcn_b[j];
    agg[idx] = fmaxf(v, 0.0f);
}

// ---------------------------------------------------------------------------

extern "C" void kernel_launch(void* const* d_in, const int* in_sizes, int n_in,
                              void* d_out, int out_size, void* d_ws, size_t ws_size,
                              hipStream_t stream) {
    const float* x     = (const float*)d_in[0];
    const int*   ei    = (const int*)d_in[1];
    const float* gcn_w = (const float*)d_in[2];
    const float* gcn_b = (const float*)d_in[3];
    const float* lin_w = (const float*)d_in[4];
    const float* lin_b = (const float*)d_in[5];
    const float* w_ih[3], *b_ih[3], *b_hh[3];
    for (int l = 0; l < 3; ++l) {
        w_ih[l] = (const float*)d_in[6 + 4 * l];
        // w_hh[l] = d_in[7+4l] unused: h0 == 0
        b_ih[l] = (const float*)d_in[8 + 4 * l];
        b_hh[l] = (const float*)d_in[9 + 4 * l];
    }

    const int N = in_sizes[0] / INF;   // 100000
    const int E = in_sizes[1] / 2;     // 1600000

    float* B0  = (float*)d_ws;                     // N*H
    float* B1  = B0 + (size_t)N * HID;             // N*H
    float* deg = B1 + (size_t)N * HID;             // N

    const int mtiles = N / 16;  // N divisible by 16

    // 1) xw = x @ gcn_w^T                 (B0)
    gemm_nt_wmma<<<dim3(mtiles, 1), 256, 0, stream>>>(x, gcn_w, nullptr, B0,
                                                      N, HID, INF);
    // 2) degree -> dinv
    hipMemsetAsync(deg, 0, (size_t)N * sizeof(float), stream);
    deg_kernel<<<(E + 255) / 256, 256, 0, stream>>>(ei, deg, E);
    dinv_kernel<<<(N + 255) / 256, 256, 0, stream>>>(deg, N);
    // 3) scatter-add normalized messages  (B1)
    hipMemsetAsync(B1, 0, (size_t)N * HID * sizeof(float), stream);
    scatter_edges<<<(E * 32 + 255) / 256, 256, 0, stream>>>(ei, B0, deg, B1, E);
    // 4) self-loop + bias + relu (in place on B1)
    selfloop_bias_relu<<<(N * HID + 255) / 256, 256, 0, stream>>>(B1, B0, deg,
                                                                  gcn_b, N);
    // 5) 3 fused LSTM steps, ping-pong B1 -> B0 -> B1 -> B0
    lstm_fused_wmma<<<mtiles, 256, 0, stream>>>(B1, w_ih[0], b_ih[0], b_hh[0], B0);
    lstm_fused_wmma<<<mtiles, 256, 0, stream>>>(B0, w_ih[1], b_ih[1], b_hh[1], B1);
    lstm_fused_wmma<<<mtiles, 256, 0, stream>>>(B1, w_ih[2], b_ih[2], b_hh[2], B0);
    // 6) head: out = h @ lin_w^T + lin_b
    gemm_nt_wmma<<<dim3(mtiles, 1), 256, 0, stream>>>(B0, lin_w, lin_b,
                                                      (float*)d_out, N, OUTF, HID);
}